// PiecewiseRationalQuadraticTransform_23158463660809
// MI455X (gfx1250) — compile-verified
//
#include <hip/hip_runtime.h>
#include <stdint.h>

// ---------------------------------------------------------------------------
// Piecewise rational-quadratic spline transform (neural spline flow), fp32.
//   inputs : (R=8192, D=2048) fp32
//   params : (D, 23) fp32   [8 widths | 8 heights | 7 derivs]
//   out    : (R, D) fp32  followed by  logdet : (R, D) fp32  in d_out.
//
// Memory-bound (≈192 MB @ 23.3 TB/s ⇒ ~8 µs). No GEMM -> no WMMA.
// CDNA5 path: per-column spline tables precomputed once into d_ws, then each
// block TDM-loads its 27x64 table tile into LDS (tensor_load_to_lds +
// s_wait_tensorcnt), and streams inputs/outputs with nontemporal b128 ops.
// ---------------------------------------------------------------------------

#define NUM_BINS 8
#define TAIL     3.0f
#define MIN_W    0.001f
#define MIN_H    0.001f
#define MIN_D    0.001f

#define NPARAM   (NUM_BINS * 3 - 1)   // 23
#define NFIELD   27                   // cw[9] | ch[9] | derivs[9]
#define CTILE    64                   // columns per block tile

typedef unsigned int u32;
typedef unsigned long long u64;
typedef __attribute__((ext_vector_type(4))) float v4f;
typedef __attribute__((ext_vector_type(4))) u32   u32x4;
typedef __attribute__((ext_vector_type(8))) int   i32x8;
typedef __attribute__((ext_vector_type(4))) int   i32x4;

__device__ __forceinline__ float softplus_stable(float x) {
    // log(1 + exp(x)) == max(x,0) + log1p(exp(-|x|))
    return fmaxf(x, 0.0f) + log1pf(expf(-fabsf(x)));
}

// ---------------------------------------------------------------------------
// Kernel 1: bake per-column spline tables into d_ws, field-major:
//   table[f * D + c],  f in [0,27):  0..8 = cw knots, 9..17 = ch knots,
//                                    18..26 = derivatives.
// ---------------------------------------------------------------------------
__global__ __launch_bounds__(256)
void rqs_precompute_tables(const float* __restrict__ params,
                           float* __restrict__ table, int D) {
    int c = blockIdx.x * blockDim.x + threadIdx.x;
    if (c >= D) return;
    const float* p = params + (size_t)c * NPARAM;

    float uw[NUM_BINS], uh[NUM_BINS];
#pragma unroll
    for (int i = 0; i < NUM_BINS; ++i) uw[i] = p[i];
#pragma unroll
    for (int i = 0; i < NUM_BINS; ++i) uh[i] = p[NUM_BINS + i];

    // ---- widths: MIN_W + (1 - MIN_W*K) * softmax(uw); knots cw[0..8] ----
    {
        float m = uw[0];
#pragma unroll
        for (int i = 1; i < NUM_BINS; ++i) m = fmaxf(m, uw[i]);
        float e[NUM_BINS], s = 0.0f;
#pragma unroll
        for (int i = 0; i < NUM_BINS; ++i) { e[i] = expf(uw[i] - m); s += e[i]; }
        float inv = 1.0f / s;
        float cs = 0.0f;
        table[0 * D + c] = -TAIL;                      // cw[0] forced exact
#pragma unroll
        for (int k = 1; k <= NUM_BINS; ++k) {
            float w = MIN_W + (1.0f - MIN_W * NUM_BINS) * e[k - 1] * inv;
            cs += w;
            table[k * D + c] = 2.0f * TAIL * cs - TAIL;
        }
        table[NUM_BINS * D + c] = TAIL;                // cw[8] forced exact
    }

    // ---- heights: same recipe; knots ch[0..8] at fields 9..17 ----
    {
        float m = uh[0];
#pragma unroll
        for (int i = 1; i < NUM_BINS; ++i) m = fmaxf(m, uh[i]);
        float e[NUM_BINS], s = 0.0f;
#pragma unroll
        for (int i = 0; i < NUM_BINS; ++i) { e[i] = expf(uh[i] - m); s += e[i]; }
        float inv = 1.0f / s;
        float cs = 0.0f;
        table[(9 + 0) * D + c] = -TAIL;
#pragma unroll
        for (int k = 1; k <= NUM_BINS; ++k) {
            float h = MIN_H + (1.0f - MIN_H * NUM_BINS) * e[k - 1] * inv;
            cs += h;
            table[(9 + k) * D + c] = 2.0f * TAIL * cs - TAIL;
        }
        table[(9 + NUM_BINS) * D + c] = TAIL;
    }

    // ---- derivatives: MIN_D + softplus(pad(ud, const)) at fields 18..26 ----
    {
        const float cpad = logf(expf(1.0f - MIN_D) - 1.0f);
        table[(18 + 0) * D + c] = MIN_D + softplus_stable(cpad);
#pragma unroll
        for (int i = 0; i < NUM_BINS - 1; ++i)
            table[(18 + 1 + i) * D + c] = MIN_D + softplus_stable(p[2 * NUM_BINS + i]);
        table[(18 + NUM_BINS) * D + c] = MIN_D + softplus_stable(cpad);
    }
}

// ---------------------------------------------------------------------------
// Kernel 2: streaming transform. Block = 256 threads = 8 waves, covers a
// 64-column tile. Table tile (27 x 64 fp32 = 6912 B) is DMA'd into LDS by the
// Tensor Data Mover, then threads stream 16 rows x 64 cols per iteration with
// b128 nontemporal loads/stores (each thread owns 4 adjacent columns).
// ---------------------------------------------------------------------------
__global__ __launch_bounds__(256)
void rqs_transform(const float* __restrict__ in,
                   const float* __restrict__ table,
                   float* __restrict__ out,
                   float* __restrict__ logdet,
                   int D, int R) {
    __shared__ float tab[NFIELD * CTILE];   // [field][col], conflict-free

    const int c0  = blockIdx.x * CTILE;
    const int tid = threadIdx.x;

#if __has_builtin(__builtin_amdgcn_tensor_load_to_lds)
    if (tid == 0) {
        // ---- Tensor DMA descriptor (D#): 2D tensor [27 rows x D cols],
        // row stride D, tile [27 x 64] starting at column c0, dword data. ----
        u64 gaddr = (u64)(uintptr_t)(table + c0);          // tile start
        u32 laddr = (u32)(uintptr_t)(&tab[0]);             // LDS byte offset

        u32x4 g0;
        g0.x = 1u;                                         // count=1 (valid D#)
        g0.y = laddr;                                      // lds_addr
        g0.z = (u32)(gaddr & 0xFFFFFFFFu);                 // global_addr[31:0]
        g0.w = (u32)((gaddr >> 32) & 0x01FFFFFFu)          // global_addr[56:32]
             | (2u << 30);                                 // type=2 ("image")

        const u32 td0 = (u32)D;        // tensor_dim0 (cols)
        const u32 td1 = NFIELD;        // tensor_dim1 (rows)
        const u64 t0s = (u64)D;        // tensor_dim0_stride

        i32x8 g1;
        g1[0] = (int)(2u << 16);                                   // data_size=4B
        g1[1] = (int)((td0 & 0xFFFFu) << 16);                      // tensor_dim0 lo
        g1[2] = (int)(((td0 >> 16) & 0xFFFFu)                      // tensor_dim0 hi
                    | ((td1 & 0xFFFFu) << 16));                    // tensor_dim1 lo
        g1[3] = (int)(((td1 >> 16) & 0xFFFFu)                      // tensor_dim1 hi
                    | ((u32)CTILE << 16));                         // tile_dim0=64
        g1[4] = (int)(u32)NFIELD;                                  // tile_dim1=27, tile_dim2=0
        g1[5] = (int)(u32)(t0s & 0xFFFFFFFFu);                     // dim0_stride lo
        g1[6] = (int)(u32)((t0s >> 32) & 0xFFFFu);                 // dim0_stride hi
        g1[7] = 0;

        i32x4 g2 = {0, 0, 0, 0};
        i32x4 g3 = {0, 0, 0, 0};
#if defined(__clang_major__) && (__clang_major__ >= 23)
        // clang-23 / therock-10.0 lane: 6-arg form
        //   (uint32x4, int32x8, int32x4, int32x4, int32x8, i32 cpol)
        i32x8 g4 = {0, 0, 0, 0, 0, 0, 0, 0};
        __builtin_amdgcn_tensor_load_to_lds(g0, g1, g2, g3, g4, 0);
#else
        // ROCm 7.2 / clang-22 lane: 5-arg form
        __builtin_amdgcn_tensor_load_to_lds(g0, g1, g2, g3, 0);
#endif
        __builtin_amdgcn_s_wait_tensorcnt(0);
    }
#else
    for (int i = tid; i < NFIELD * CTILE; i += 256)
        tab[i] = table[(size_t)(i >> 6) * D + c0 + (i & (CTILE - 1))];
#endif
    __syncthreads();

    const int cl  = (tid & 15) * 4;        // column offset within tile (0..60)
    const int rIn = tid >> 4;              // row lane within 16-row group
    const int nRowGroups = R >> 4;

    for (int rg = blockIdx.y; rg < nRowGroups; rg += gridDim.y) {
        const int    r    = (rg << 4) + rIn;
        const size_t base = (size_t)r * D + c0 + cl;

        v4f xin = __builtin_nontemporal_load((const v4f*)(in + base));
        v4f o, ld;

#pragma unroll
        for (int j = 0; j < 4; ++j) {
            const int col = cl + j;
            const float x0 = xin[j];
            const bool inside = (x0 >= -TAIL) && (x0 <= TAIL);
            const float x = fminf(fmaxf(x0, -TAIL), TAIL);

            // bin index: count interior knots <= x (matches sum(x>=cw)-1 clip)
            int idx = 0;
#pragma unroll
            for (int k = 1; k <= NUM_BINS - 1; ++k)
                idx += (x >= tab[k * CTILE + col]) ? 1 : 0;

            const float icw = tab[idx * CTILE + col];
            const float iw  = tab[(idx + 1) * CTILE + col] - icw;
            const float ich = tab[(9 + idx) * CTILE + col];
            const float ih  = tab[(9 + idx + 1) * CTILE + col] - ich;
            const float dlo = tab[(18 + idx) * CTILE + col];
            const float dhi = tab[(18 + idx + 1) * CTILE + col];

            const float delta = ih / iw;
            const float theta = (x - icw) / iw;
            const float t1m   = theta * (1.0f - theta);
            const float th2   = theta * theta;
            const float omt   = 1.0f - theta;

            const float num = ih * (delta * th2 + dlo * t1m);
            const float den = delta + (dlo + dhi - 2.0f * delta) * t1m;
            const float o_in = ich + num / den;

            const float dnum = delta * delta *
                               (dhi * th2 + 2.0f * delta * t1m + dlo * omt * omt);
            const float l_in = logf(dnum) - 2.0f * logf(den);

            o[j]  = inside ? o_in : x0;
            ld[j] = inside ? l_in : 0.0f;
        }

        __builtin_nontemporal_store(o,  (v4f*)(out + base));
        __builtin_nontemporal_store(ld, (v4f*)(logdet + base));
    }
}

// ---------------------------------------------------------------------------
extern "C" void kernel_launch(void* const* d_in, const int* in_sizes, int n_in,
                              void* d_out, int out_size, void* d_ws, size_t ws_size,
                              hipStream_t stream) {
    const float* inputs = (const float*)d_in[0];   // (R, D) fp32
    const float* params = (const float*)d_in[1];   // (D, 23) fp32

    const int D = in_sizes[1] / NPARAM;            // 2048
    const int R = in_sizes[0] / D;                 // 8192

    float* out    = (float*)d_out;
    float* logdet = out + (size_t)R * D;
    float* table  = (float*)d_ws;                  // 27 * D fp32 = 216 KB

    // 1) bake per-column spline tables once
    rqs_precompute_tables<<<(D + 255) / 256, 256, 0, stream>>>(params, table, D);

    // 2) streaming transform: 32 column tiles x 128 row blocks, 256 thr each
    dim3 grid(D / CTILE, 128);
    rqs_transform<<<grid, 256, 0, stream>>>(inputs, table, out, logdet, D, R);
}